// HetGAT_NoSem_76682346102829
// MI455X (gfx1250) — compile-verified
//
#include <hip/hip_runtime.h>
#include <hip/hip_bf16.h>

typedef float v2f __attribute__((ext_vector_type(2)));
typedef float v8f __attribute__((ext_vector_type(8)));

// ---------------------------------------------------------------------------
// WMMA f32 GEMM: C[M,N] = act(A[M,K] @ W[K,N] + bias[N])
// One 16x16 output tile per wave. Requires M%16==0, N%16==0, K%4==0.
// A-frag (16x4 f32): lane L<16 -> row M=L, v0=K0,v1=K1 ; lane L>=16 -> row M=L-16, v0=K2,v1=K3
// B-frag (4x16 f32): lane L<16 -> col N=L, v0=K0,v1=K1 ; lane L>=16 -> col N=L-16, v0=K2,v1=K3
// C/D (16x16 f32):   VGPR r, lanes 0-15: M=r,  N=lane ; lanes 16-31: M=r+8, N=lane-16
// act: 0 = none, 1 = relu
// ---------------------------------------------------------------------------
__global__ __launch_bounds__(256) void gemm_wmma_f32(
    const float* __restrict__ A, const float* __restrict__ W,
    const float* __restrict__ bias, float* __restrict__ C,
    int M, int K, int N, int act) {
  const int lane = threadIdx.x & 31;
  const int wave = threadIdx.x >> 5;
  const int tilesN = N >> 4;
  const int tile = blockIdx.x * (blockDim.x >> 5) + wave;
  const int totalTiles = (M >> 4) * tilesN;
  if (tile >= totalTiles) return;           // wave-uniform: EXEC stays all-ones
  const int rowTile = tile / tilesN;
  const int colTile = tile - rowTile * tilesN;
  const int row0 = rowTile << 4;
  const int col0 = colTile << 4;
  const int half = lane >> 4;               // 0 or 1
  const int l15  = lane & 15;

  const float* __restrict__ Arow = A + (size_t)(row0 + l15) * K;
  const float* __restrict__ Wcol = W + (size_t)(col0 + l15);

  v8f acc = {};
  for (int k0 = 0; k0 < K; k0 += 4) {
    const int ka = k0 + 2 * half;
    v2f a, b;
    a.x = Arow[ka];
    a.y = Arow[ka + 1];
    b.x = Wcol[(size_t)ka * N];
    b.y = Wcol[(size_t)(ka + 1) * N];
    acc = __builtin_amdgcn_wmma_f32_16x16x4_f32(
        /*neg_a=*/false, a, /*neg_b=*/false, b,
        /*c_mod=*/(short)0, acc, /*reuse_a=*/false, /*reuse_b=*/false);
  }

  const int col = col0 + l15;
  const float bv = bias ? bias[col] : 0.0f;
#pragma unroll
  for (int r = 0; r < 8; ++r) {
    const int row = row0 + r + 8 * half;
    float v = acc[r] + bv;
    if (act == 1) v = v > 0.0f ? v : 0.0f;
    C[(size_t)row * N + col] = v;
  }
}

// ---------------------------------------------------------------------------
// Per-source-node prep: d1 = t[n]·a1, d2 = t[n]·a2, w2 = exp(leaky(d1+d2)).
// Writes sSrc[n]=d1, div[n]=w2, agg[n,:]=w2*t[n,:].  One wave (32 lanes) per node, H=64.
// ---------------------------------------------------------------------------
__global__ __launch_bounds__(256) void node_prep_src(
    const float* __restrict__ t, const float* __restrict__ a1,
    const float* __restrict__ a2, float* __restrict__ sSrc,
    float* __restrict__ divv, float* __restrict__ agg, int Nn) {
  const int lane = threadIdx.x & 31;
  const int node = (int)((blockIdx.x * (unsigned)blockDim.x + threadIdx.x) >> 5);
  if (node >= Nn) return;
  const float* __restrict__ tr = t + (size_t)node * 64;
  const float t0 = tr[lane], t1 = tr[lane + 32];
  float d1 = t0 * a1[lane] + t1 * a1[lane + 32];
  float d2 = t0 * a2[lane] + t1 * a2[lane + 32];
#pragma unroll
  for (int off = 16; off > 0; off >>= 1) {
    d1 += __shfl_xor(d1, off, 32);
    d2 += __shfl_xor(d2, off, 32);
  }
  float z = d1 + d2;
  z = z > 0.0f ? z : 0.2f * z;
  const float w2 = __expf(z);
  agg[(size_t)node * 64 + lane]      = w2 * t0;
  agg[(size_t)node * 64 + lane + 32] = w2 * t1;
  if (lane == 0) { sSrc[node] = d1; divv[node] = w2; }
}

// Per-destination-node prep: sDst[n] = t[n]·a2.  One wave per node.
__global__ __launch_bounds__(256) void node_prep_dst(
    const float* __restrict__ t, const float* __restrict__ a2,
    float* __restrict__ sDst, int Nn) {
  const int lane = threadIdx.x & 31;
  const int node = (int)((blockIdx.x * (unsigned)blockDim.x + threadIdx.x) >> 5);
  if (node >= Nn) return;
  const float* __restrict__ tr = t + (size_t)node * 64;
  float d = tr[lane] * a2[lane] + tr[lane + 32] * a2[lane + 32];
#pragma unroll
  for (int off = 16; off > 0; off >>= 1) d += __shfl_xor(d, off, 32);
  if (lane == 0) sDst[node] = d;
}

// ---------------------------------------------------------------------------
// Edge aggregation: 64 lanes per edge.
//   w1 = exp(leaky(sSrc[s] + sDst[t]))
//   agg[s,:] += w1 * hDst[t,:]   (atomics)
//   div[s]   += w1               (lane 0)
// ---------------------------------------------------------------------------
__global__ __launch_bounds__(256) void edge_agg(
    const int* __restrict__ ei, const float* __restrict__ sSrc,
    const float* __restrict__ sDst, const float* __restrict__ hDst,
    float* __restrict__ agg, float* __restrict__ divv, int E) {
  const long long gid = (long long)blockIdx.x * blockDim.x + threadIdx.x;
  const int e = (int)(gid >> 6);
  const int k = (int)(gid & 63);
  if (e >= E) return;
  const int s = ei[e];
  const int t = ei[E + e];
  float z = sSrc[s] + sDst[t];
  z = z > 0.0f ? z : 0.2f * z;
  const float w1 = __expf(z);
  atomicAdd(&agg[(size_t)s * 64 + k], w1 * hDst[(size_t)t * 64 + k]);
  if (k == 0) atomicAdd(&divv[s], w1);
}

// In-place: agg = elu(agg / div)
__global__ __launch_bounds__(256) void finalize_elu(
    float* __restrict__ agg, const float* __restrict__ divv, int Nn) {
  const int gid = blockIdx.x * blockDim.x + threadIdx.x;
  if (gid >= Nn * 64) return;
  const int n = gid >> 6;
  const float v = agg[gid] / divv[n];
  agg[gid] = v > 0.0f ? v : (__expf(v) - 1.0f);
}

// ---------------------------------------------------------------------------
extern "C" void kernel_launch(void* const* d_in, const int* in_sizes, int n_in,
                              void* d_out, int out_size, void* d_ws, size_t ws_size,
                              hipStream_t stream) {
  (void)in_sizes; (void)n_in; (void)out_size; (void)ws_size;
  const float* x_P     = (const float*)d_in[0];
  const float* x_A     = (const float*)d_in[1];
  const int*   ei_pa   = (const int*)  d_in[2];
  const int*   ei_ap   = (const int*)  d_in[3];
  const float* fc1_P_w = (const float*)d_in[4];
  const float* fc1_P_b = (const float*)d_in[5];
  const float* fc1_A_w = (const float*)d_in[6];
  const float* fc1_A_b = (const float*)d_in[7];
  const float* fcs_w   = (const float*)d_in[8];
  const float* fcs_b   = (const float*)d_in[9];
  const float* a1_pa   = (const float*)d_in[10];
  const float* a2_pa   = (const float*)d_in[11];
  const float* a1_ap   = (const float*)d_in[12];
  const float* a2_ap   = (const float*)d_in[13];
  const float* fc2_w   = (const float*)d_in[14];
  const float* fc2_b   = (const float*)d_in[15];
  float* out = (float*)d_out;

  const int NP = 200000, NA = 200000, E = 2000000, H = 64, OUT = 32, HOP = 2;

  float* ws = (float*)d_ws;
  float* hP    = ws; ws += (size_t)NP * H;   // current P features / agg buffer
  float* hA    = ws; ws += (size_t)NA * H;   // current A features / agg buffer
  float* tP    = ws; ws += (size_t)NP * H;   // transformed P features
  float* tA    = ws; ws += (size_t)NA * H;   // transformed A features
  float* divP  = ws; ws += NP;
  float* divA  = ws; ws += NA;
  float* sSrcP = ws; ws += NP;
  float* sSrcA = ws; ws += NA;
  float* sDstP = ws; ws += NP;
  float* sDstA = ws; ws += NA;

  auto launch_gemm = [&](const float* A, const float* W, const float* b, float* C,
                         int M, int K, int N, int act) {
    const int tiles  = (M / 16) * (N / 16);
    const int blocks = (tiles + 7) / 8;     // 8 waves (256 threads) per block
    gemm_wmma_f32<<<blocks, 256, 0, stream>>>(A, W, b, C, M, K, N, act);
  };

  // Input projections with ReLU
  launch_gemm(x_P, fc1_P_w, fc1_P_b, hP, NP, 128, H, 1);
  launch_gemm(x_A, fc1_A_w, fc1_A_b, hA, NA, 64,  H, 1);

  const int nodeBlocksP = (NP + 7) / 8;     // wave-per-node, 8 nodes per block
  const int nodeBlocksA = (NA + 7) / 8;
  const long long edgeThreads = (long long)E * 64;
  const int edgeBlocks = (int)((edgeThreads + 255) / 256);

  for (int i = 0; i < HOP; ++i) {
    const float* Wi = fcs_w + (size_t)i * H * H;
    const float* bi = fcs_b + (size_t)i * H;
    launch_gemm(hP, Wi, bi, tP, NP, H, H, 0);
    launch_gemm(hA, Wi, bi, tA, NA, H, H, 0);

    // Edge type (P -> A): src = P, dst = A
    node_prep_src<<<nodeBlocksP, 256, 0, stream>>>(tP, a1_pa + i * H, a2_pa + i * H,
                                                   sSrcP, divP, hP, NP);
    node_prep_dst<<<nodeBlocksA, 256, 0, stream>>>(tA, a2_pa + i * H, sDstA, NA);
    // Edge type (A -> P): src = A, dst = P
    node_prep_src<<<nodeBlocksA, 256, 0, stream>>>(tA, a1_ap + i * H, a2_ap + i * H,
                                                   sSrcA, divA, hA, NA);
    node_prep_dst<<<nodeBlocksP, 256, 0, stream>>>(tP, a2_ap + i * H, sDstP, NP);

    edge_agg<<<edgeBlocks, 256, 0, stream>>>(ei_pa, sSrcP, sDstA, tA, hP, divP, E);
    edge_agg<<<edgeBlocks, 256, 0, stream>>>(ei_ap, sSrcA, sDstP, tP, hA, divA, E);

    finalize_elu<<<(NP * H + 255) / 256, 256, 0, stream>>>(hP, divP, NP);
    finalize_elu<<<(NA * H + 255) / 256, 256, 0, stream>>>(hA, divA, NA);
  }

  // Output head: out = hP @ fc2_w + fc2_b
  launch_gemm(hP, fc2_w, fc2_b, out, NP, 64, OUT, 0);
}